// DecoderOnlyAttention_65369402245615
// MI455X (gfx1250) — compile-verified
//
#include <hip/hip_runtime.h>
#include <hip/hip_bf16.h>

#define HIDDEN   4096
#define NHEADS   32
#define KVHEADS  8
#define HDIM     128
#define NBATCH   32
#define MAXSEQ   2048
#define BLKSZ    128
#define NBLK     16
#define GRP      4
#define NQ       (NHEADS*HDIM)    // 4096
#define NKV      (KVHEADS*HDIM)   // 1024
#define KSPLIT   8
#define KCHUNK   (HIDDEN/KSPLIT)  // 512

typedef __attribute__((ext_vector_type(2))) float v2f;
typedef __attribute__((ext_vector_type(8))) float v8f;

// D(16x16 f32) = A(16x4 f32) * B(4x16 f32) + C
__device__ __forceinline__ v8f wmma4(v2f a, v2f b, v8f c) {
  return __builtin_amdgcn_wmma_f32_16x16x4_f32(
      /*neg_a=*/false, a, /*neg_b=*/false, b,
      /*c_mod=*/(short)0, c, /*reuse_a=*/false, /*reuse_b=*/false);
}

// ---------------------------------------------------------------------------
// Kernel A: fused QKV projection + RoPE, split-K in-workgroup.
// One block per (head, 16-col tile-pair): cols [h*128+16j,+16) and
// [h*128+64+16j,+16) so rotate_half partners live in the same block.
// 8 waves each accumulate a disjoint 512-K chunk -> LDS -> reduce + RoPE.
// Blocks 0..127 -> q, 128..159 -> k, 160..191 -> v.  192 blocks x 8 waves.
// ---------------------------------------------------------------------------
__global__ __launch_bounds__(256) void qkv_rope(
    const float* __restrict__ X,      // 32 x 4096
    const float* __restrict__ wq,     // 4096 x 4096
    const float* __restrict__ wk,     // 4096 x 1024
    const float* __restrict__ wv,     // 4096 x 1024
    const float* __restrict__ cosT,   // 32 x 128
    const float* __restrict__ sinT,   // 32 x 128
    float* __restrict__ qbuf,         // 32 x 4096  (rotated)
    float* __restrict__ knew,         // 32 x 8 x 128 (rotated)
    float* __restrict__ vnew)         // 32 x 8 x 128
{
  __shared__ float lds[KSPLIT * 1024];   // 8 chunks x (32 rows x 32 cols) = 32 KB

  const int w    = blockIdx.x;           // output tile-pair id
  const int ch   = threadIdx.x >> 5;     // K-chunk owned by this wave
  const int lane = threadIdx.x & 31;
  const int lhi  = lane >> 4;
  const int llo  = lane & 15;

  const float* W; int N, head, jj, kind;
  if (w < 128)      { kind = 0; head = w >> 2;         jj = w & 3;         W = wq; N = NQ;  }
  else if (w < 160) { kind = 1; head = (w - 128) >> 2; jj = (w - 128) & 3; W = wk; N = NKV; }
  else              { kind = 2; head = (w - 160) >> 2; jj = (w - 160) & 3; W = wv; N = NKV; }

  const int colA = head * HDIM + jj * 16 + llo;   // dim in [0,64)
  const int colB = colA + 64;                     // rotate_half partner

  v8f acc00 = {}, acc01 = {}, acc10 = {}, acc11 = {};
  const float* xr0 = X + llo * HIDDEN;            // batches 0-15
  const float* xr1 = X + (llo + 16) * HIDDEN;     // batches 16-31

  const int kbeg = ch * KCHUNK;
#pragma unroll 4
  for (int k0 = kbeg; k0 < kbeg + KCHUNK; k0 += 4) {
    const int kb = k0 + 2 * lhi;
    v2f a0 = *(const v2f*)(xr0 + kb);
    v2f a1 = *(const v2f*)(xr1 + kb);
    const float* wr = W + (size_t)kb * N;
    v2f b0, b1;
    b0.x = wr[colA];      b1.x = wr[colB];
    b0.y = wr[N + colA];  b1.y = wr[N + colB];
    acc00 = wmma4(a0, b0, acc00);   // B reused across both M tiles:
    acc10 = wmma4(a1, b0, acc10);   // each weight element read exactly once
    acc01 = wmma4(a0, b1, acc01);
    acc11 = wmma4(a1, b1, acc11);
  }

  // ---- dump partial C tiles to LDS:  lds[ch][batch(32)][col(32)] ----------
#pragma unroll
  for (int mt = 0; mt < 2; ++mt) {
#pragma unroll
    for (int r = 0; r < 8; ++r) {
      const int b = r + 8 * lhi + 16 * mt;
      lds[ch * 1024 + b * 32 + llo]      = (mt == 0) ? acc00[r] : acc10[r];
      lds[ch * 1024 + b * 32 + 16 + llo] = (mt == 0) ? acc01[r] : acc11[r];
    }
  }
  __syncthreads();

  // ---- reduce 8 partials + fused RoPE epilogue ----------------------------
  for (int p = threadIdx.x; p < 512; p += 256) {
    const int row = p >> 4;        // batch
    const int c   = p & 15;        // col within 16-wide tile
    float sA = 0.f, sB = 0.f;
#pragma unroll
    for (int q = 0; q < KSPLIT; ++q) {
      sA += lds[q * 1024 + row * 32 + c];
      sB += lds[q * 1024 + row * 32 + 16 + c];
    }
    const int dA = jj * 16 + c;    // dim in [0,64); cos[d]==cos[d+64]
    if (kind == 2) {
      vnew[(row * KVHEADS + head) * HDIM + dA]      = sA;
      vnew[(row * KVHEADS + head) * HDIM + dA + 64] = sB;
    } else {
      const float cv = cosT[row * HDIM + dA];
      const float sv = sinT[row * HDIM + dA];
      const float oA = sA * cv - sB * sv;      // d < 64 : rot = -q[d+64]
      const float oB = sB * cv + sA * sv;      // d >= 64: rot = +q[d-64]
      if (kind == 0) {
        qbuf[row * NQ + head * HDIM + dA]      = oA;
        qbuf[row * NQ + head * HDIM + dA + 64] = oB;
      } else {
        knew[(row * KVHEADS + head) * HDIM + dA]      = oA;
        knew[(row * KVHEADS + head) * HDIM + dA + 64] = oB;
      }
    }
  }
}

// ---------------------------------------------------------------------------
// Kernel B: paged attention. One workgroup per (batch, kv_head).
// Scores (4 q-heads padded to M=16) -> LDS, workgroup softmax, probs @ V.
// Current position reads knew/vnew instead of the (unmodified) cache.
// ---------------------------------------------------------------------------
__global__ __launch_bounds__(256) void paged_attn(
    const float* __restrict__ qbuf,
    const float* __restrict__ knew,
    const float* __restrict__ vnew,
    const float* __restrict__ kcache,   // (512, 8, 128, 128)
    const float* __restrict__ vcache,
    const int*   __restrict__ block_tables,  // (32, 16)
    const int*   __restrict__ seq_positions, // (32,)
    float* __restrict__ ctxbuf)              // 32 x 4096
{
  __shared__ float probs[GRP * MAXSEQ];   // 32 KB
  __shared__ float red[GRP * 64];

  const int b    = blockIdx.x >> 3;
  const int h    = blockIdx.x & 7;
  const int tid  = threadIdx.x;
  const int wid  = tid >> 5;
  const int lane = tid & 31, lhi = lane >> 4, llo = lane & 15;

  const int pos  = seq_positions[b];
  const int spad = ((pos + 1) + 15) & ~15;     // compute only what the mask keeps
  const int* bt  = block_tables + b * NBLK;
  const float* qrow  = qbuf + b * NQ + h * GRP * HDIM;
  const float* knewp = knew + (b * KVHEADS + h) * HDIM;
  const float* vnewp = vnew + (b * KVHEADS + h) * HDIM;
  const float scale = 0.08838834764831845f;    // 1/sqrt(128)

  // -------- scores: q(16x128, 4 valid rows) @ K^T, 16 keys per WMMA tile ----
  for (int s0 = wid * 16; s0 < spad; s0 += 128) {
    const int s = s0 + llo;                    // this lane's key position
    const float* kptr;
    if (s == pos) {
      kptr = knewp;                            // freshly-projected key
    } else {
      const int blk = bt[s >> 7];
      kptr = kcache + (((size_t)blk * KVHEADS + h) * BLKSZ + (s & (BLKSZ - 1))) * HDIM;
    }
    v8f c = {};
#pragma unroll 8
    for (int k0 = 0; k0 < HDIM; k0 += 4) {
      const int kb = k0 + 2 * lhi;
      v2f a, bb;
      a.x = (llo < GRP) ? qrow[llo * HDIM + kb]     : 0.f;
      a.y = (llo < GRP) ? qrow[llo * HDIM + kb + 1] : 0.f;
      bb.x = kptr[kb];
      bb.y = kptr[kb + 1];
      c = wmma4(a, bb, c);
    }
    if (lhi == 0) {
#pragma unroll
      for (int r = 0; r < GRP; ++r)
        probs[r * MAXSEQ + s0 + llo] = c[r] * scale;
    }
  }
  __syncthreads();

  // -------- softmax over s in [0, pos]; 64 threads per q-head row ----------
  const int row = tid >> 6;
  const int t   = tid & 63;
  float mx = -1e30f;
  for (int s = t; s < spad; s += 64) {
    const float v = (s <= pos) ? probs[row * MAXSEQ + s] : -1e9f;
    mx = fmaxf(mx, v);
  }
  red[row * 64 + t] = mx;
  __syncthreads();
  mx = -1e30f;
  for (int i = 0; i < 64; ++i) mx = fmaxf(mx, red[row * 64 + i]);
  __syncthreads();
  float sm = 0.f;
  for (int s = t; s < spad; s += 64) {
    const float p = (s <= pos) ? __expf(probs[row * MAXSEQ + s] - mx) : 0.f;
    probs[row * MAXSEQ + s] = p;
    sm += p;
  }
  red[row * 64 + t] = sm;
  __syncthreads();
  sm = 0.f;
  for (int i = 0; i < 64; ++i) sm += red[row * 64 + i];
  const float inv = 1.f / sm;
  __syncthreads();
  for (int s = t; s < spad; s += 64)
    probs[row * MAXSEQ + s] *= inv;
  __syncthreads();

  // -------- ctx = probs(4 x spad) @ V(spad x 128); 16 dims per wave --------
  const int dcol = wid * 16 + llo;
  v8f c = {};
  for (int k0 = 0; k0 < spad; k0 += 4) {
    const int kb = k0 + 2 * lhi;
    v2f a, bb;
    a.x = (llo < GRP) ? probs[llo * MAXSEQ + kb]     : 0.f;
    a.y = (llo < GRP) ? probs[llo * MAXSEQ + kb + 1] : 0.f;
    const float* v0;
    const float* v1;
    if (kb == pos) { v0 = vnewp; }
    else { const int blk = bt[kb >> 7];
           v0 = vcache + (((size_t)blk * KVHEADS + h) * BLKSZ + (kb & 127)) * HDIM; }
    const int kb1 = kb + 1;
    if (kb1 == pos) { v1 = vnewp; }
    else { const int blk = bt[kb1 >> 7];
           v1 = vcache + (((size_t)blk * KVHEADS + h) * BLKSZ + (kb1 & 127)) * HDIM; }
    bb.x = v0[dcol];
    bb.y = v1[dcol];
    c = wmma4(a, bb, c);
  }
  if (lhi == 0) {
#pragma unroll
    for (int r = 0; r < GRP; ++r)
      ctxbuf[b * NQ + (h * GRP + r) * HDIM + dcol] = c[r];
  }
}

// ---------------------------------------------------------------------------
// Kernel C: output projection (32 x 4096) @ wo(4096 x 4096) -> d_out.
// One block per 16-col tile (256 blocks); 8 waves split K 8 ways, reduce in
// LDS. Both M tiles reuse the B fragment -> each wo element read once.
// ---------------------------------------------------------------------------
__global__ __launch_bounds__(256) void out_proj(
    const float* __restrict__ ctxbuf,
    const float* __restrict__ wo,
    float* __restrict__ out)
{
  __shared__ float lds[KSPLIT * 512];    // 8 chunks x (32 rows x 16 cols) = 16 KB

  const int ch   = threadIdx.x >> 5;
  const int lane = threadIdx.x & 31, lhi = lane >> 4, llo = lane & 15;
  const int col  = blockIdx.x * 16 + llo;

  v8f c0 = {}, c1 = {};
  const float* xr0 = ctxbuf + llo * NQ;
  const float* xr1 = ctxbuf + (llo + 16) * NQ;

  const int kbeg = ch * KCHUNK;
#pragma unroll 4
  for (int k0 = kbeg; k0 < kbeg + KCHUNK; k0 += 4) {
    const int kb = k0 + 2 * lhi;
    v2f a0 = *(const v2f*)(xr0 + kb);
    v2f a1 = *(const v2f*)(xr1 + kb);
    v2f bb;
    bb.x = wo[(size_t)kb * HIDDEN + col];
    bb.y = wo[(size_t)(kb + 1) * HIDDEN + col];
    c0 = wmma4(a0, bb, c0);
    c1 = wmma4(a1, bb, c1);
  }

#pragma unroll
  for (int r = 0; r < 8; ++r) {
    const int b0 = r + 8 * lhi;
    lds[ch * 512 + b0 * 16 + llo]        = c0[r];
    lds[ch * 512 + (b0 + 16) * 16 + llo] = c1[r];
  }
  __syncthreads();

  for (int p = threadIdx.x; p < 512; p += 256) {
    const int row = p >> 4;
    const int c   = p & 15;
    float s = 0.f;
#pragma unroll
    for (int q = 0; q < KSPLIT; ++q)
      s += lds[q * 512 + row * 16 + c];
    out[(size_t)row * HIDDEN + blockIdx.x * 16 + c] = s;
  }
}

// ---------------------------------------------------------------------------
extern "C" void kernel_launch(void* const* d_in, const int* in_sizes, int n_in,
                              void* d_out, int out_size, void* d_ws, size_t ws_size,
                              hipStream_t stream) {
  const float* hidden = (const float*)d_in[0];
  // d_in[1] = attn_mask (unused; mask derived from seq_positions)
  const int*   seqpos = (const int*)d_in[2];
  const float* kcache = (const float*)d_in[3];
  const float* vcache = (const float*)d_in[4];
  const int*   btab   = (const int*)d_in[5];
  const float* cosT   = (const float*)d_in[6];
  const float* sinT   = (const float*)d_in[7];
  const float* wq     = (const float*)d_in[8];
  const float* wk     = (const float*)d_in[9];
  const float* wv     = (const float*)d_in[10];
  const float* wo     = (const float*)d_in[11];
  float* out = (float*)d_out;

  float* ws   = (float*)d_ws;
  float* qbuf = ws;                       // 32*4096
  float* knew = qbuf + NBATCH * NQ;       // 32*1024
  float* vnew = knew + NBATCH * NKV;      // 32*1024
  float* ctxb = vnew + NBATCH * NKV;      // 32*4096

  qkv_rope  <<<192,              256, 0, stream>>>(hidden, wq, wk, wv, cosT, sinT,
                                                   qbuf, knew, vnew);
  paged_attn<<<NBATCH * KVHEADS, 256, 0, stream>>>(qbuf, knew, vnew, kcache, vcache,
                                                   btab, seqpos, ctxb);
  out_proj  <<<256,              256, 0, stream>>>(ctxb, wo, out);
}